// Nonlinearity_35940286333411
// MI455X (gfx1250) — compile-verified
//
#include <hip/hip_runtime.h>

typedef __attribute__((ext_vector_type(2))) float v2f;
typedef __attribute__((ext_vector_type(4))) float v4f;
typedef __attribute__((ext_vector_type(8))) float v8f;

constexpr int  kC  = 512;
constexpr long kCC = (long)kC * kC;

// Static LDS layout (floats):
//   [0,   392) : cg, padded to [M][n][8] (m-slots >= K1 are zero)
//   [392, 520) : x1 tile, padded to [16][8] (m-slots >= K1 are zero)
//   [520, ...) : per-wave store-staging, 256*NM floats per wave (NM<=7)
constexpr int kCgOff = 0;
constexpr int kX1Off = 392;
constexpr int kStOff = 520;
constexpr int kSmemFloats = kStOff + 8 * 256 * 7;  // 14856 floats = 59424 B

struct TPArgs {
  const float* x[4];    // x0..x3
  const float* cg[23];  // cg in _triples() order
  float*       out;     // concatenated outputs (float32)
};

// One wave computes one 16x16 (i,j) tile of p[i,j,M] for one triple via
// V_WMMA_F32_16X16X4_F32:  D = A(16xK1) * B(K1x16) + C  per output row M.
//   A[i][m] = x1[i][m]          (staged zero-padded in LDS)
//   B[m][j] = sum_n cg[M][n][m] * x2[j][n]   (cg staged zero-padded in LDS)
template <int L1, int L2, int LL>
__device__ __forceinline__ void tile_tp(const float* __restrict__ x1,
                                        const float* __restrict__ x2,
                                        const float* __restrict__ cg,
                                        float* __restrict__ out,
                                        float* __restrict__ smem,
                                        int i0, int jt, int wave, int lane,
                                        int tid) {
  constexpr int K1 = 2 * L1 + 1;   // A K-dim (1,3,5,7)
  constexpr int K2 = 2 * L2 + 1;   // contracted n-dim for B build
  constexpr int NM = 2 * LL + 1;   // number of output rows M

  // ---- Phase 1: cooperative zero-padded staging (once per block) ----
  for (int p = tid; p < NM * K2 * 8; p += 256) {
    const int mn = p >> 3;          // M*K2 + n
    const int mm = p & 7;           // padded m slot
    smem[kCgOff + p] = (mm < K1) ? cg[mn * K1 + mm] : 0.0f;
  }
  for (int p = tid; p < 16 * 8; p += 256) {
    const int row = p >> 3;
    const int mm  = p & 7;
    smem[kX1Off + p] = (mm < K1) ? x1[(i0 + row) * K1 + mm] : 0.0f;
  }
  __syncthreads();

  const int half = lane >> 4;       // 0: lanes 0-15, 1: lanes 16-31
  const int lid  = lane & 15;
  const int jcol = jt * 16 + lid;   // B-matrix column (j)

  // ---- A fragments from LDS (16x4 f32 layout), guard-free ----
  const v2f* ax   = (const v2f*)(smem + kX1Off + lid * 8);
  const v2f  a_lo = ax[half];       // K = 2*half, 2*half+1
  const v2f  a_hi = ax[half + 2];   // K = 2*half+4, 2*half+5

  // x2 row for this lane's j (tiny, L2/L0-resident)
  float x2v[K2];
#pragma unroll
  for (int n = 0; n < K2; ++n) x2v[n] = x2[jcol * K2 + n];

  v8f acc[NM];
#pragma unroll
  for (int M = 0; M < NM; ++M) acc[M] = v8f{0, 0, 0, 0, 0, 0, 0, 0};

#pragma unroll
  for (int M = 0; M < NM; ++M) {
    float t0 = 0.f, t1 = 0.f, t2 = 0.f, t3 = 0.f;
#pragma unroll
    for (int n = 0; n < K2; ++n) {
      const v2f* cgp  = (const v2f*)(smem + kCgOff + (M * K2 + n) * 8);
      const float xv  = x2v[n];
      const v2f  c_lo = cgp[half];
      t0 += c_lo.x * xv;
      t1 += c_lo.y * xv;
      if constexpr (K1 > 4) {
        const v2f c_hi = cgp[half + 2];
        t2 += c_hi.x * xv;
        t3 += c_hi.y * xv;
      }
    }
    v2f b_lo; b_lo.x = t0; b_lo.y = t1;
    acc[M] = __builtin_amdgcn_wmma_f32_16x16x4_f32(
        false, a_lo, false, b_lo, (short)0, acc[M], false, false);
    if constexpr (K1 > 4) {
      v2f b_hi; b_hi.x = t2; b_hi.y = t3;
      acc[M] = __builtin_amdgcn_wmma_f32_16x16x4_f32(
          false, a_hi, false, b_hi, (short)0, acc[M], false, false);
    }
  }

  // ---- Phase 3: per-wave LDS transpose -> coalesced b128 global stores ----
  // Staging layout matches global row segments: lst[row][j_local*NM + M].
  float* lst = smem + kStOff + wave * (256 * NM);
  const int rbase = half * 8;       // C/D layout: VGPR v -> row v + 8*half
#pragma unroll
  for (int v = 0; v < 8; ++v) {
    float* drow = lst + ((rbase + v) * 16 + lid) * NM;
#pragma unroll
    for (int M = 0; M < NM; ++M) drow[M] = acc[M][v];
  }
  // Wave-local readback (LDS ops from one wave stay in order; compiler
  // inserts s_wait_dscnt). Each lane moves 8*NM floats = 2*NM float4s.
  const int r   = lane >> 1;        // output row 0..15
  const int seg = lane & 1;         // half-row segment
  const v4f* src  = (const v4f*)(lst + (r * 16 + seg * 8) * NM);
  v4f*       gdst = (v4f*)(out + ((long)(i0 + r) * kC + (long)jt * 16) * NM) +
                    seg * 2 * NM;
#pragma unroll
  for (int c = 0; c < 2 * NM; ++c) gdst[c] = src[c];
}

__global__ __launch_bounds__(256, 1) void cg_tp_kernel(TPArgs args) {
  __shared__ float smem[kSmemFloats];

  const int tid  = (int)threadIdx.x;
  const int lane = tid & 31;
  const int wave = tid >> 5;
  // 8 consecutive tiles per block share one i-tile (8 | 32).
  const int i0 = ((int)blockIdx.x >> 2) << 4;            // i-tile * 16
  const int jt = (((int)blockIdx.x & 3) << 3) + wave;    // j-tile 0..31

  switch (blockIdx.y) {  // triple id, uniform per block -> EXEC all-1s at WMMA
#define TP_CASE(T, A, B, L, OFF)                                           \
  case T:                                                                  \
    tile_tp<A, B, L>(args.x[A], args.x[B], args.cg[T],                     \
                     args.out + (long)(OFF) * kCC, smem, i0, jt, wave,     \
                     lane, tid);                                           \
    break;
    TP_CASE(0,  0, 0, 0,  0)   // (0,0,0) -> out0 block 0
    TP_CASE(1,  0, 1, 1,  4)   // (0,1,1) -> out1 block 0
    TP_CASE(2,  0, 2, 2, 22)   // (0,2,2) -> out2 block 0
    TP_CASE(3,  0, 3, 3, 57)   // (0,3,3) -> out3 block 0
    TP_CASE(4,  1, 1, 0,  1)   // (1,1,0) -> out0 block 1
    TP_CASE(5,  1, 1, 1,  7)   // (1,1,1) -> out1 block 1
    TP_CASE(6,  1, 1, 2, 27)   // (1,1,2) -> out2 block 1
    TP_CASE(7,  1, 2, 1, 10)   // (1,2,1) -> out1 block 2
    TP_CASE(8,  1, 2, 2, 32)   // (1,2,2) -> out2 block 2
    TP_CASE(9,  1, 2, 3, 64)   // (1,2,3) -> out3 block 1
    TP_CASE(10, 1, 3, 2, 37)   // (1,3,2) -> out2 block 3
    TP_CASE(11, 1, 3, 3, 71)   // (1,3,3) -> out3 block 2
    TP_CASE(12, 2, 2, 0,  2)   // (2,2,0) -> out0 block 2
    TP_CASE(13, 2, 2, 1, 13)   // (2,2,1) -> out1 block 3
    TP_CASE(14, 2, 2, 2, 42)   // (2,2,2) -> out2 block 4
    TP_CASE(15, 2, 2, 3, 78)   // (2,2,3) -> out3 block 3
    TP_CASE(16, 2, 3, 1, 16)   // (2,3,1) -> out1 block 4
    TP_CASE(17, 2, 3, 2, 47)   // (2,3,2) -> out2 block 5
    TP_CASE(18, 2, 3, 3, 85)   // (2,3,3) -> out3 block 4
    TP_CASE(19, 3, 3, 0,  3)   // (3,3,0) -> out0 block 3
    TP_CASE(20, 3, 3, 1, 19)   // (3,3,1) -> out1 block 5
    TP_CASE(21, 3, 3, 2, 52)   // (3,3,2) -> out2 block 6
    TP_CASE(22, 3, 3, 3, 92)   // (3,3,3) -> out3 block 5
#undef TP_CASE
    default: break;
  }
}

extern "C" void kernel_launch(void* const* d_in, const int* in_sizes, int n_in,
                              void* d_out, int out_size, void* d_ws,
                              size_t ws_size, hipStream_t stream) {
  (void)in_sizes; (void)n_in; (void)out_size; (void)d_ws; (void)ws_size;

  TPArgs args;
  for (int l = 0; l < 4; ++l) args.x[l] = (const float*)d_in[l];
  for (int t = 0; t < 23; ++t) args.cg[t] = (const float*)d_in[4 + t];
  args.out = (float*)d_out;

  // 23 triples; per triple 32x32 tiles of 16x16 = 1024 waves = 128 blocks of
  // 8 wave32s. grid.y selects the triple (block-uniform).
  dim3 grid(128, 23, 1);
  dim3 block(256, 1, 1);
  cg_tp_kernel<<<grid, block, 0, stream>>>(args);
}